// S4DKernel_46686294507696
// MI455X (gfx1250) — compile-verified
//
#include <hip/hip_runtime.h>
#include <hip/hip_bf16.h>

#define HD 512      // hidden dim H
#define LD 4096     // sequence length L
#define RD 32       // dt_rank R
#define N2 16       // state dim / 2

typedef __attribute__((ext_vector_type(2))) float v2f;
typedef __attribute__((ext_vector_type(8))) float v8f;

struct cplx { float x, y; };
__device__ __forceinline__ cplx cmul(cplx a, cplx b) {
    return { a.x * b.x - a.y * b.y, a.x * b.y + a.y * b.x };
}

// ---------------------------------------------------------------------------
// Kernel 1: transpose u[L,H] -> uT[H,L] via LDS tiles (coalesced both ways)
// ---------------------------------------------------------------------------
__global__ void transpose_u_kernel(const float* __restrict__ u, float* __restrict__ uT) {
    __shared__ float tile[32][33];
    const int hb = blockIdx.x * 32;        // h tile base
    const int lb = blockIdx.y * 32;        // l tile base
    const int tx = threadIdx.x & 31;
    const int ty = threadIdx.x >> 5;       // 8 rows per step
    for (int i = ty; i < 32; i += 8)
        tile[i][tx] = u[(size_t)(lb + i) * HD + hb + tx];
    __syncthreads();
    for (int i = ty; i < 32; i += 8)
        uT[(size_t)(hb + i) * LD + lb + tx] = tile[tx][i];
}

// ---------------------------------------------------------------------------
// Kernel 2: dt path with fp32 WMMA (V_WMMA_F32_16X16X4_F32).
//   dt_u[l,r] = sum_h u[l,h] * xproj_w[r,h]          (GEMM1, K=512)
//   dt[l,h]   = softplus(sum_r dt_u[l,r]*dt_w[h,r] + dt_b[h])   (GEMM2, K=32)
// Stored transposed: dtT[h*L + l].
// Block: 256 threads (8 waves). grid = (L/16, H/128).
// Fragment layouts (ISA 7.12.2, 32-bit):
//   A 16x4 : lane&15 = M; (lane>>4)? K={2,3} : K={0,1} in v0,v1
//   B 4x16 : lane&15 = N; (lane>>4)? K={2,3} : K={0,1} in v0,v1  (mirror of A)
//   C/D 16x16: vgpr i -> M = i + 8*(lane>>4), N = lane&15
// ---------------------------------------------------------------------------
__global__ void dt_kernel(const float* __restrict__ u,
                          const float* __restrict__ xproj_w,   // [R,H]
                          const float* __restrict__ dt_w,      // [H,R]
                          const float* __restrict__ dt_b,      // [H]
                          float* __restrict__ dtT) {           // [H,L]
    __shared__ float u_lds[16 * HD];     // 32 KB: u tile [16][512]
    __shared__ float du_lds[16 * RD];    // dt_u tile [16][32]

    const int tid  = threadIdx.x;
    const int wave = tid >> 5;
    const int lane = tid & 31;
    const int l0   = blockIdx.x * 16;

    for (int i = tid; i < 16 * HD; i += 256) {
        int m = i >> 9, k = i & (HD - 1);
        u_lds[i] = u[(size_t)(l0 + m) * HD + k];
    }
    __syncthreads();

    const int m  = lane & 15;            // M row (A/C) or N col (B)
    const int kb = (lane >> 4) << 1;     // K sub-offset: 0 or 2

    if (wave < 2) {
        // GEMM1: this wave computes dt_u columns r = wave*16 .. wave*16+15
        v8f acc = {};
        const int r = wave * 16 + m;     // B fragment N index
        for (int k = 0; k < HD; k += 4) {
            v2f a, b;
            a.x = u_lds[m * HD + k + kb];
            a.y = u_lds[m * HD + k + kb + 1];
            b.x = xproj_w[r * HD + k + kb];
            b.y = xproj_w[r * HD + k + kb + 1];
            acc = __builtin_amdgcn_wmma_f32_16x16x4_f32(
                      false, a, false, b, (short)0, acc, false, false);
        }
        const int col = wave * 16 + (lane & 15);
        #pragma unroll
        for (int i = 0; i < 8; i++) {
            int row = i + ((lane >> 4) << 3);
            du_lds[row * RD + col] = acc[i];
        }
    }
    __syncthreads();

    // GEMM2: each wave handles h tile h0..h0+15
    const int h0 = blockIdx.y * 128 + wave * 16;
    const int hc = h0 + m;               // B fragment N index
    v8f acc = {};
    #pragma unroll
    for (int k = 0; k < RD; k += 4) {
        v2f a, b;
        a.x = du_lds[m * RD + k + kb];
        a.y = du_lds[m * RD + k + kb + 1];
        b.x = dt_w[hc * RD + k + kb];
        b.y = dt_w[hc * RD + k + kb + 1];
        acc = __builtin_amdgcn_wmma_f32_16x16x4_f32(
                  false, a, false, b, (short)0, acc, false, false);
    }
    const float bias = dt_b[hc];
    #pragma unroll
    for (int i = 0; i < 8; i++) {
        int lrow = l0 + i + ((lane >> 4) << 3);
        float x  = acc[i] + bias;
        float sp = (x > 20.0f) ? x : log1pf(__expf(x));   // softplus
        dtT[(size_t)hc * LD + lrow] = sp;
    }
}

// ---------------------------------------------------------------------------
// Kernel 3: ZOH discretization + chunked parallel scan + output.
// One block per h. 256 threads = 16 chunks (c) x 16 states (n).
// Chunk length = 256. Two-pass recompute scan; carry combine in LDS.
// ---------------------------------------------------------------------------
__global__ void scan_kernel(const float* __restrict__ uT,      // [H,L]
                            const float* __restrict__ dtT,     // [H,L]
                            const float* __restrict__ A_log,   // [H,N2]
                            const float* __restrict__ A_im,    // [H,N2]
                            const float* __restrict__ B_param, // [H,N2,2]
                            const float* __restrict__ C_param, // [H,N2,2]
                            const float* __restrict__ Dp,      // [H]
                            float* __restrict__ y) {           // [L,H]
    const int h   = blockIdx.x;
    const int tid = threadIdx.x;
    const int n   = tid & 15;
    const int c   = tid >> 4;
    const int CL  = LD / 16;             // 256

    __shared__ float u_l[LD];
    __shared__ float dt_l[LD];
    __shared__ float ApX[16][16], ApY[16][16];
    __shared__ float HeX[16][16], HeY[16][16];
    __shared__ float CaX[16][16], CaY[16][16];

    for (int i = tid; i < LD; i += 256) {
        __builtin_prefetch(&uT[(size_t)h * LD + i + 256], 0, 0);
        u_l[i]  = uT[(size_t)h * LD + i];
        dt_l[i] = dtT[(size_t)h * LD + i];
    }

    const int hn = h * N2 + n;
    const float Ar = -__expf(A_log[hn]);
    const float Ai = A_im[hn];
    const cplx Bc = { B_param[hn * 2 + 0], B_param[hn * 2 + 1] };
    const cplx Cc = { C_param[hn * 2 + 0], C_param[hn * 2 + 1] };
    __syncthreads();

    const int lbase = c * CL;

    // ---- Pass 1: per-chunk (prod A_bar, h_end) with carry-in 0 ----
    cplx Ap = { 1.f, 0.f }, hacc = { 0.f, 0.f };
    for (int j = 0; j < CL; j++) {
        const float dt = dt_l[lbase + j];
        const float uu = u_l[lbase + j];
        const float zr = dt * Ar, zi = dt * Ai;
        const float er = __expf(zr);
        const float cz = __cosf(zi), sz = __sinf(zi);
        const cplx Abar = { er * cz, er * sz };
        const float zm2 = zr * zr + zi * zi;
        cplx Bdiv;
        if (zm2 < 1e-8f) {
            Bdiv = { 1.f + 0.5f * zr, 0.5f * zi };
        } else {
            const cplx num = { Abar.x - 1.f, Abar.y };
            const float inv = 1.f / zm2;
            Bdiv = { (num.x * zr + num.y * zi) * inv,
                     (num.y * zr - num.x * zi) * inv };
        }
        cplx ub = cmul(Bdiv, Bc);
        ub.x *= uu; ub.y *= uu;
        hacc = cmul(Abar, hacc);
        hacc.x += ub.x; hacc.y += ub.y;
        Ap = cmul(Abar, Ap);
    }
    ApX[c][n] = Ap.x;  ApY[c][n] = Ap.y;
    HeX[c][n] = hacc.x; HeY[c][n] = hacc.y;
    __syncthreads();

    // ---- Carry combine across 16 chunks (16 threads, one per state) ----
    if (tid < 16) {
        cplx carry = { 0.f, 0.f };
        for (int cc = 0; cc < 16; cc++) {
            CaX[cc][tid] = carry.x; CaY[cc][tid] = carry.y;
            const cplx ap = { ApX[cc][tid], ApY[cc][tid] };
            const cplx he = { HeX[cc][tid], HeY[cc][tid] };
            carry = cmul(ap, carry);
            carry.x += he.x; carry.y += he.y;
        }
    }
    __syncthreads();

    // ---- Pass 2: replay with carry-in, reduce over states, write y ----
    cplx hst = { CaX[c][n], CaY[c][n] };
    const float Dv = Dp[h];
    for (int j = 0; j < CL; j++) {
        const float dt = dt_l[lbase + j];
        const float uu = u_l[lbase + j];
        const float zr = dt * Ar, zi = dt * Ai;
        const float er = __expf(zr);
        const float cz = __cosf(zi), sz = __sinf(zi);
        const cplx Abar = { er * cz, er * sz };
        const float zm2 = zr * zr + zi * zi;
        cplx Bdiv;
        if (zm2 < 1e-8f) {
            Bdiv = { 1.f + 0.5f * zr, 0.5f * zi };
        } else {
            const cplx num = { Abar.x - 1.f, Abar.y };
            const float inv = 1.f / zm2;
            Bdiv = { (num.x * zr + num.y * zi) * inv,
                     (num.y * zr - num.x * zi) * inv };
        }
        cplx ub = cmul(Bdiv, Bc);
        ub.x *= uu; ub.y *= uu;
        hst = cmul(Abar, hst);
        hst.x += ub.x; hst.y += ub.y;

        // Re(h * Cc), reduced over the 16 state-lanes (wave32 butterflies)
        float contrib = hst.x * Cc.x - hst.y * Cc.y;
        contrib += __shfl_xor(contrib, 1, 32);
        contrib += __shfl_xor(contrib, 2, 32);
        contrib += __shfl_xor(contrib, 4, 32);
        contrib += __shfl_xor(contrib, 8, 32);
        if (n == 0)
            y[(size_t)(lbase + j) * HD + h] = contrib + uu * Dv;
    }
}

// ---------------------------------------------------------------------------
extern "C" void kernel_launch(void* const* d_in, const int* in_sizes, int n_in,
                              void* d_out, int out_size, void* d_ws, size_t ws_size,
                              hipStream_t stream) {
    const float* u       = (const float*)d_in[0];
    const float* A_log   = (const float*)d_in[1];
    const float* A_im    = (const float*)d_in[2];
    const float* B_param = (const float*)d_in[3];
    const float* C_param = (const float*)d_in[4];
    const float* Dp      = (const float*)d_in[5];
    const float* dt_w    = (const float*)d_in[6];
    const float* dt_b    = (const float*)d_in[7];
    const float* xproj_w = (const float*)d_in[8];

    float* uT  = (float*)d_ws;                    // [H, L]  8 MB
    float* dtT = uT + (size_t)HD * LD;            // [H, L]  8 MB

    transpose_u_kernel<<<dim3(HD / 32, LD / 32), 256, 0, stream>>>(u, uT);
    dt_kernel<<<dim3(LD / 16, HD / 128), 256, 0, stream>>>(u, xproj_w, dt_w, dt_b, dtT);
    scan_kernel<<<HD, 256, 0, stream>>>(uT, dtT, A_log, A_im, B_param, C_param,
                                        Dp, (float*)d_out);
}